// Rope_71201967833256
// MI455X (gfx1250) — compile-verified
//
#include <hip/hip_runtime.h>

// RoPE: x (16, 8192, 128) f32, token_position (16, 8192) int.
// out[..., 2k]   = cos(a)*x[2k] - sin(a)*x[2k+1]
// out[..., 2k+1] = sin(a)*x[2k] + cos(a)*x[2k+1],  a = pos * theta^(-k/64)
//
// Memory-bound: ~129 MiB total traffic @ 23.3 TB/s => ~5.5 us roofline.
//  - one wave32 per 128-float row; lane l owns floats [4l,4l+4) (pairs 2l,2l+1)
//  - per-thread inverse-frequency constants via one v_exp_f32, hoisted
//  - sin/cos in revolution domain -> v_sin_f32 / v_cos_f32 (valid range ±0.5)
//  - gfx1250 async path: GLOBAL_LOAD_ASYNC_TO_LDS_B128 double-buffers whole
//    512B rows through wave-private LDS slots (ASYNCcnt, zero VGPR pressure),
//    s_wait_asynccnt(1) overlaps copy of row i+1 with compute of row i.
//  - NT store for the write-once output stream.

typedef float v4f __attribute__((ext_vector_type(4)));
typedef int   v4i __attribute__((vector_size(16)));   // builtin's pointee type

#define ROPE_LOG2_THETA 13.287712379549449f   /* log2(10000) */
#define ROPE_LOG2_2PI   2.651496129472319f    /* log2(2*pi)  */

#if __has_builtin(__builtin_amdgcn_global_load_async_to_lds_b128) && \
    __has_builtin(__builtin_amdgcn_s_wait_asynccnt)
#define ROPE_ASYNC 1
#else
#define ROPE_ASYNC 0
#endif

__global__ __launch_bounds__(256) void rope_f32_kernel(
    const float* __restrict__ x,
    const int*   __restrict__ pos,
    float*       __restrict__ out,
    int nrows)
{
    const int lane          = threadIdx.x & 31;
    const int waveInBlock   = threadIdx.x >> 5;
    const int wavesPerBlock = blockDim.x >> 5;          // 8
    const int rowStride     = gridDim.x * wavesPerBlock;
    int row = blockIdx.x * wavesPerBlock + waveInBlock;

    // Pair indices owned by this lane: k0 = 2*lane, k1 = 2*lane + 1.
    // c_k = theta^(-k/64) / (2*pi) == exp2(-k*log2(theta)/64 - log2(2*pi)),
    // computed once per thread and reused for every row.
    const float k0 = (float)(2 * lane);
    const float k1 = k0 + 1.0f;
    const float c0 = __builtin_exp2f(__builtin_fmaf(k0, -ROPE_LOG2_THETA * (1.0f / 64.0f), -ROPE_LOG2_2PI));
    const float c1 = __builtin_exp2f(__builtin_fmaf(k1, -ROPE_LOG2_THETA * (1.0f / 64.0f), -ROPE_LOG2_2PI));

#if ROPE_ASYNC
    // Wave-private double buffer: 2 stages x 8 waves x 32 lanes x 16B = 8 KB.
    __shared__ v4f stage[2][8][32];

    if (row < nrows) {
        __builtin_amdgcn_global_load_async_to_lds_b128(
            (v4i*)(x + (size_t)row * 128 + 4 * lane),
            (v4i*)&stage[0][waveInBlock][lane], 0, 0);
    }
    int p = 0;
    for (; row < nrows; row += rowStride) {
        const int nrow = row + rowStride;
        if (nrow < nrows) {
            // Prefetch next row into the other stage, keep 1 copy in flight.
            __builtin_amdgcn_global_load_async_to_lds_b128(
                (v4i*)(x + (size_t)nrow * 128 + 4 * lane),
                (v4i*)&stage[p ^ 1][waveInBlock][lane], 0, 0);
            __builtin_amdgcn_s_wait_asynccnt(1);   // stage p is now resident
        } else {
            __builtin_amdgcn_s_wait_asynccnt(0);
        }
        __asm__ volatile("" ::: "memory");          // keep LDS read below the wait

        v4f   v  = stage[p][waveInBlock][lane];     // ds_load_b128
        float pf = (float)pos[row];

        float t0 = pf * c0;  t0 -= __builtin_rintf(t0);
        float t1 = pf * c1;  t1 -= __builtin_rintf(t1);
        float s0 = __builtin_amdgcn_sinf(t0);
        float g0 = __builtin_amdgcn_cosf(t0);
        float s1 = __builtin_amdgcn_sinf(t1);
        float g1 = __builtin_amdgcn_cosf(t1);

        v4f o;
        o.x = __builtin_fmaf(g0, v.x, -s0 * v.y);
        o.y = __builtin_fmaf(s0, v.x,  g0 * v.y);
        o.z = __builtin_fmaf(g1, v.z, -s1 * v.w);
        o.w = __builtin_fmaf(s1, v.z,  g1 * v.w);

        __builtin_nontemporal_store(o, reinterpret_cast<v4f*>(out + (size_t)row * 128) + lane);
        p ^= 1;
    }
#else
    // Fallback: direct b128 loads + explicit prefetch of the next strided row.
    for (; row < nrows; row += rowStride) {
        const v4f* __restrict__ src =
            reinterpret_cast<const v4f*>(x + (size_t)row * 128) + lane;
        v4f* __restrict__ dst =
            reinterpret_cast<v4f*>(out + (size_t)row * 128) + lane;

        {
            int nrow = row + rowStride;
            if (nrow < nrows) {
                const char* pf = reinterpret_cast<const char*>(
                    reinterpret_cast<const v4f*>(x + (size_t)nrow * 128) + lane);
                __builtin_prefetch(pf, 0, 0);       // global_prefetch_b8
            }
        }

        v4f   v  = __builtin_nontemporal_load(src);
        float pf = (float)pos[row];

        float t0 = pf * c0;  t0 -= __builtin_rintf(t0);
        float t1 = pf * c1;  t1 -= __builtin_rintf(t1);
        float s0 = __builtin_amdgcn_sinf(t0);
        float g0 = __builtin_amdgcn_cosf(t0);
        float s1 = __builtin_amdgcn_sinf(t1);
        float g1 = __builtin_amdgcn_cosf(t1);

        v4f o;
        o.x = __builtin_fmaf(g0, v.x, -s0 * v.y);
        o.y = __builtin_fmaf(s0, v.x,  g0 * v.y);
        o.z = __builtin_fmaf(g1, v.z, -s1 * v.w);
        o.w = __builtin_fmaf(s1, v.z,  g1 * v.w);

        __builtin_nontemporal_store(o, dst);
    }
#endif
}

extern "C" void kernel_launch(void* const* d_in, const int* in_sizes, int n_in,
                              void* d_out, int out_size, void* d_ws, size_t ws_size,
                              hipStream_t stream) {
    (void)n_in; (void)d_ws; (void)ws_size; (void)out_size;

    const float* x   = (const float*)d_in[0];
    const int*   pos = (const int*)d_in[1];
    float*       out = (float*)d_out;

    const int nrows = in_sizes[0] / 128;            // 16 * 8192 = 131072 rows

    const int block = 256;                          // 8 wave32 per block
    const int wavesPerBlock = block / 32;
    int grid = 4096;                                // ~4 rows per wave
    int maxGrid = (nrows + wavesPerBlock - 1) / wavesPerBlock;
    if (grid > maxGrid) grid = maxGrid;

    rope_f32_kernel<<<grid, block, 0, stream>>>(x, pos, out, nrows);
}